// NeuralStateMapper_64630667870425
// MI455X (gfx1250) — compile-verified
//
#include <hip/hip_runtime.h>
#include <hip/hip_bf16.h>
#include <math.h>

typedef __attribute__((ext_vector_type(2))) float v2f;
typedef __attribute__((ext_vector_type(8))) float v8f;

#define D_STATE 16
#define D_INNER 128
#define NB 16
#define NL 8192
#define NM (NB * NL)        /* 131072 rows (b*L+l) */
#define XZ_LD 256           /* xz row stride: [u_pre | z], delta overwrites u_pre */
#define DBC_LD 48           /* dbc padded row stride: dt[0:4] B[4:20] C[20:36] pad[36:48] */

/* workspace layout in floats */
#define OFF_XZ   ((size_t)0)
#define OFF_U    ((size_t)NM * XZ_LD)                 /* 33,554,432 */
#define OFF_DBC  (OFF_U + (size_t)NM * D_INNER)       /* +16,777,216 */
#define OFF_YSUM (OFF_DBC + (size_t)NM * DBC_LD)      /* +6,291,456 */

/* ---------------- K1: xz = x @ W_in  (131072x64 @ 64x256), f32 WMMA -------
 * W_in staged in LDS TRANSPOSED: sWt[col][k], stride 66 (bank-rotated).
 * B fragment (rows kr,kr+1 at column col) = one contiguous 8B ds_load_b64. */
#define K1_TLD 66
__global__ __launch_bounds__(256) void k_gemm_in(const float* __restrict__ x,
                                                 const float* __restrict__ Win,
                                                 float* __restrict__ xz) {
  __shared__ float sWt[128 * K1_TLD]; /* ~33KB: half of W_in's cols, transposed */
  const int tid  = threadIdx.x;
  const int wave = tid >> 5;
  const int lane = tid & 31;
  const int m    = lane & 15;
  const int kb   = (lane >> 4) << 1;           /* k base: 0 or 2 */
  const long mbase = ((long)blockIdx.x * 8 + wave) * 16;

  /* A fragments: full 16x64 tile, 2 f32 per lane per k-step, reused 16x */
  v2f a[16];
#pragma unroll
  for (int ks = 0; ks < 16; ++ks)
    a[ks] = *(const v2f*)(x + (mbase + m) * 64 + ks * 4 + kb);

  for (int half = 0; half < 2; ++half) {
    /* coalesced global read (consecutive cols of one k-row), transposed LDS write */
    for (int i = tid; i < 64 * 128; i += 256) {
      const int c = i & 127, k = i >> 7;
      sWt[c * K1_TLD + k] = Win[k * 256 + half * 128 + c];
    }
    __syncthreads();
#pragma unroll
    for (int nt = 0; nt < 8; ++nt) {
      v8f acc = {};
      const int col = nt * 16 + m;
#pragma unroll
      for (int ks = 0; ks < 16; ++ks) {
        const v2f bf = *(const v2f*)(sWt + col * K1_TLD + ks * 4 + kb);
        acc = __builtin_amdgcn_wmma_f32_16x16x4_f32(false, a[ks], false, bf,
                                                    (short)0, acc, false, false);
      }
      const int mr = (lane >> 4) * 8;
#pragma unroll
      for (int v = 0; v < 8; ++v)
        xz[(mbase + mr + v) * XZ_LD + half * 128 + nt * 16 + m] = acc[v];
    }
    __syncthreads();
  }
}

/* ---------------- K2: depthwise causal conv(4) + SiLU -------------------- */
__global__ __launch_bounds__(256) void k_conv_silu(const float* __restrict__ xz,
                                                   const float* __restrict__ conv_w,
                                                   const float* __restrict__ conv_b,
                                                   float* __restrict__ u) {
  const long idx = (long)blockIdx.x * 256 + threadIdx.x; /* over NM*128 */
  const int  d   = (int)(idx & 127);
  const long row = idx >> 7;                              /* b*L + l */
  const int  l   = (int)(row & (NL - 1));
  float acc = conv_b[d];
#pragma unroll
  for (int j = 0; j < 4; ++j) {
    const int lj = l - 3 + j;
    const float xv = (lj >= 0) ? xz[(row - 3 + j) * XZ_LD + d] : 0.0f;
    acc = fmaf(xv, conv_w[d * 4 + j], acc);
  }
  u[row * (long)D_INNER + d] = acc / (1.0f + __expf(-acc));
}

/* ---------------- K3: dbc = u @ W_x (K=128, N=36 padded to 48), WMMA ------
 * W_x transposed+padded in LDS: sWt[col][k], stride 130. */
#define K3_TLD 130
__global__ __launch_bounds__(256) void k_gemm_xproj(const float* __restrict__ u,
                                                    const float* __restrict__ Wx,
                                                    float* __restrict__ dbc) {
  __shared__ float sWt[DBC_LD * K3_TLD]; /* ~25KB */
  const int tid = threadIdx.x, wave = tid >> 5, lane = tid & 31;
  const int m = lane & 15, kb = (lane >> 4) << 1;
  const long mbase = ((long)blockIdx.x * 8 + wave) * 16;
  for (int i = tid; i < 128 * DBC_LD; i += 256) {
    const int k = i / DBC_LD, c = i % DBC_LD;
    sWt[c * K3_TLD + k] = (c < 36) ? Wx[k * 36 + c] : 0.0f;
  }
  v2f a[32];
#pragma unroll
  for (int ks = 0; ks < 32; ++ks)
    a[ks] = *(const v2f*)(u + (mbase + m) * D_INNER + ks * 4 + kb);
  __syncthreads();
#pragma unroll
  for (int nt = 0; nt < 3; ++nt) {
    v8f acc = {};
    const int col = nt * 16 + m;
#pragma unroll
    for (int ks = 0; ks < 32; ++ks) {
      const v2f bf = *(const v2f*)(sWt + col * K3_TLD + ks * 4 + kb);
      acc = __builtin_amdgcn_wmma_f32_16x16x4_f32(false, a[ks], false, bf,
                                                  (short)0, acc, false, false);
    }
    const int mr = (lane >> 4) * 8;
#pragma unroll
    for (int v = 0; v < 8; ++v)
      dbc[(mbase + mr + v) * DBC_LD + col] = acc[v];
  }
}

/* ------- K4: delta = softplus(dt @ W_dt + b_dt), one WMMA per tile --------
 * W_dt (4x128) transposed in LDS: sWdt[col][k], stride 4 -> b frag = b64 load */
__global__ __launch_bounds__(256) void k_delta(const float* __restrict__ dbc,
                                               const float* __restrict__ Wdt,
                                               const float* __restrict__ bdt,
                                               float* __restrict__ xz) {
  __shared__ float sWdt[128 * 4]; /* 2KB transposed */
  const int tid = threadIdx.x, wave = tid >> 5, lane = tid & 31;
  const int m = lane & 15, kb = (lane >> 4) << 1;
  const long mbase = ((long)blockIdx.x * 8 + wave) * 16;
  for (int i = tid; i < 512; i += 256) {
    const int c = i & 127, k = i >> 7;
    sWdt[c * 4 + k] = Wdt[k * 128 + c];
  }
  const v2f a = *(const v2f*)(dbc + (mbase + m) * DBC_LD + kb); /* dt cols 0..3 */
  __syncthreads();
  const int mr = (lane >> 4) * 8;
#pragma unroll
  for (int nt = 0; nt < 8; ++nt) {
    const int col = nt * 16 + m;
    const v2f bf = *(const v2f*)(sWdt + col * 4 + kb);
    v8f acc = {};
    acc = __builtin_amdgcn_wmma_f32_16x16x4_f32(false, a, false, bf,
                                                (short)0, acc, false, false);
    const float bias = bdt[col];
#pragma unroll
    for (int v = 0; v < 8; ++v) {
      const float t  = acc[v] + bias;
      const float sp = (t > 20.0f) ? t : log1pf(__expf(t));
      xz[(mbase + mr + v) * XZ_LD + col] = sp; /* overwrite u_pre slot */
    }
  }
}

/* ------- K5: selective scan, fused gating + Dp skip + L-reduction ---------
 * Serial over L (latency-bound): prefetch operand rows ahead of the exp chain */
__global__ __launch_bounds__(128) void k_scan(const float* __restrict__ xz,
                                              const float* __restrict__ u,
                                              const float* __restrict__ dbc,
                                              const float* __restrict__ A_log,
                                              const float* __restrict__ Dp,
                                              float* __restrict__ y_sum) {
  const int b = blockIdx.x, d = threadIdx.x;
  float Arow[D_STATE];
#pragma unroll
  for (int n = 0; n < D_STATE; ++n) Arow[n] = -__expf(A_log[d * D_STATE + n]);
  const float dp = Dp[d];
  float h[D_STATE];
#pragma unroll
  for (int n = 0; n < D_STATE; ++n) h[n] = 0.0f;
  float acc = 0.0f;
  const long rbase = (long)b * NL;
  for (int l = 0; l < NL; ++l) {
    const long row   = rbase + l;
    if (l + 8 < NL) { /* gfx1250 global_prefetch_b8: hide latency behind exps */
      const long prow = row + 8;
      __builtin_prefetch(xz + prow * XZ_LD + d, 0, 1);
      __builtin_prefetch(u + prow * (long)D_INNER + d, 0, 1);
      __builtin_prefetch(dbc + prow * DBC_LD + 4, 0, 1);
    }
    const float delta = xz[row * XZ_LD + d];
    const float zval  = xz[row * XZ_LD + 128 + d];
    const float uval  = u[row * (long)D_INNER + d];
    const float4* bc  = (const float4*)(dbc + row * DBC_LD + 4);
    float Bv[16], Cv[16];
#pragma unroll
    for (int q = 0; q < 4; ++q) {
      const float4 t = bc[q];     /* B: dbc cols 4..19  */
      const float4 s = bc[q + 4]; /* C: dbc cols 20..35 */
      Bv[4*q] = t.x; Bv[4*q+1] = t.y; Bv[4*q+2] = t.z; Bv[4*q+3] = t.w;
      Cv[4*q] = s.x; Cv[4*q+1] = s.y; Cv[4*q+2] = s.z; Cv[4*q+3] = s.w;
    }
    const float du = delta * uval;
    float y = 0.0f;
#pragma unroll
    for (int n = 0; n < D_STATE; ++n) {
      const float dA = __expf(delta * Arow[n]);
      h[n] = fmaf(h[n], dA, du * Bv[n]);
      y = fmaf(h[n], Cv[n], y);
    }
    const float sig = 1.0f / (1.0f + __expf(-zval));
    acc = fmaf(y + uval * dp, zval * sig, acc); /* gated out, summed over L */
  }
  y_sum[b * D_INNER + d] = acc;
}

/* ------- K6: pooled = (ysum/L) @ W_out ; relu-MLP ; sigmoid --------------- */
__global__ __launch_bounds__(512) void k_head(const float* __restrict__ y_sum,
                                              const float* __restrict__ Wout,
                                              const float* __restrict__ fc1_w,
                                              const float* __restrict__ fc1_b,
                                              const float* __restrict__ fc2_w,
                                              const float* __restrict__ fc2_b,
                                              float* __restrict__ out) {
  __shared__ float pooled[16 * 64];
  __shared__ float hid[16 * 32];
  const int tid = threadIdx.x;
  for (int i = tid; i < 16 * 64; i += 512) {
    const int bb = i >> 6, j = i & 63;
    float s = 0.0f;
    for (int k = 0; k < 128; ++k) s = fmaf(y_sum[bb * 128 + k], Wout[k * 64 + j], s);
    pooled[i] = s * (1.0f / (float)NL);
  }
  __syncthreads();
  {
    const int bb = tid >> 5, j = tid & 31; /* 512 threads == 16*32 */
    float s = fc1_b[j];
    for (int k = 0; k < 64; ++k) s = fmaf(pooled[bb * 64 + k], fc1_w[k * 32 + j], s);
    hid[tid] = fmaxf(s, 0.0f);
  }
  __syncthreads();
  if (tid < 16) {
    float s = fc2_b[0];
    for (int k = 0; k < 32; ++k) s = fmaf(hid[tid * 32 + k], fc2_w[k], s);
    out[tid] = 1.0f / (1.0f + __expf(-s));
  }
}

extern "C" void kernel_launch(void* const* d_in, const int* in_sizes, int n_in,
                              void* d_out, int out_size, void* d_ws, size_t ws_size,
                              hipStream_t stream) {
  const float* x      = (const float*)d_in[0];
  const float* W_in   = (const float*)d_in[1];
  const float* conv_w = (const float*)d_in[2];
  const float* conv_b = (const float*)d_in[3];
  const float* W_x    = (const float*)d_in[4];
  const float* W_dt   = (const float*)d_in[5];
  const float* b_dt   = (const float*)d_in[6];
  const float* A_log  = (const float*)d_in[7];
  const float* Dp     = (const float*)d_in[8];
  const float* W_out  = (const float*)d_in[9];
  const float* fc1_w  = (const float*)d_in[10];
  const float* fc1_b  = (const float*)d_in[11];
  const float* fc2_w  = (const float*)d_in[12];
  const float* fc2_b  = (const float*)d_in[13];

  float* ws   = (float*)d_ws;
  float* xz   = ws + OFF_XZ;
  float* u    = ws + OFF_U;
  float* dbc  = ws + OFF_DBC;
  float* ysum = ws + OFF_YSUM;
  float* out  = (float*)d_out;

  k_gemm_in   <<<NM / 128, 256, 0, stream>>>(x, W_in, xz);
  k_conv_silu <<<(NM / 2), 256, 0, stream>>>(xz, conv_w, conv_b, u); /* NM*128/256 */
  k_gemm_xproj<<<NM / 128, 256, 0, stream>>>(u, W_x, dbc);
  k_delta     <<<NM / 128, 256, 0, stream>>>(dbc, W_dt, b_dt, xz);
  k_scan      <<<NB, 128, 0, stream>>>(xz, u, dbc, A_log, Dp, ysum);
  k_head      <<<1, 512, 0, stream>>>(ysum, W_out, fc1_w, fc1_b, fc2_w, fc2_b, out);
}